// ACM_GNN_4337916969351
// MI455X (gfx1250) — compile-verified
//
#include <hip/hip_runtime.h>
#include <hip/hip_bf16.h>

typedef __attribute__((ext_vector_type(2))) float v2f;
typedef __attribute__((ext_vector_type(8))) float v8f;

constexpr int N_NODES = 50000;
constexpr int IN_F    = 256;
constexpr int HID_F   = 128;
constexpr int OUT_F   = 40;
constexpr int N_EDGES = 800000;

// ---------------------------------------------------------------- utilities
__global__ void zero_f32(float* __restrict__ p, long long n) {
    long long i = (long long)blockIdx.x * blockDim.x + threadIdx.x;
    if (i < n) p[i] = 0.0f;
}

__global__ void degree_kernel(const int* __restrict__ rows, float* __restrict__ deg, int E) {
    int e = blockIdx.x * blockDim.x + threadIdx.x;
    if (e < E)
        __hip_atomic_fetch_add(&deg[rows[e]], 1.0f, __ATOMIC_RELAXED, __HIP_MEMORY_SCOPE_AGENT);
}

__global__ void dinv_kernel(float* __restrict__ d, int n) {
    int i = blockIdx.x * blockDim.x + threadIdx.x;
    if (i < n) d[i] = 1.0f / (1.0f + d[i]);   // rowsum = 1 + deg > 0 always
}

// -------------------------------------------------- fp32 WMMA 16x16x4 GEMM
// Computes O? = X @ W? for three weight matrices (selected by blockIdx.y).
// One wave produces one 16x16 tile of the output.  N_NODES % 16 == 0, so no
// row guard.  Tail-column masking is wave-uniform, so the common case runs a
// clean load->wmma loop; EXEC stays all-ones across every v_wmma.
// FIN/FOUT are compile-time so B loads use immediate offsets.
template<int FIN, int FOUT>
__global__ void gemm3_wmma(const float* __restrict__ X,
                           const float* __restrict__ Wa, const float* __restrict__ Wb,
                           const float* __restrict__ Wc,
                           float* __restrict__ Oa, float* __restrict__ Ob,
                           float* __restrict__ Oc)
{
    const float* W = (blockIdx.y == 0) ? Wa : (blockIdx.y == 1 ? Wb : Wc);
    float*       O = (blockIdx.y == 0) ? Oa : (blockIdx.y == 1 ? Ob : Oc);

    const int wave = threadIdx.x >> 5;
    const int lane = threadIdx.x & 31;
    const int l16  = lane & 15;
    const int hi   = lane >> 4;                 // 0: K pair {k,k+1}; 1: {k+2,k+3}
    const int rowBase = blockIdx.x << 4;
    const int m    = rowBase + l16;             // A-matrix row for this lane
    const int ncol = (wave << 4) + l16;         // B/D column for this lane

    const float* xrow = X + (size_t)m * FIN + (hi << 1);
    v8f acc = {};

    if (((wave << 4) + 16) <= FOUT) {
        // -------- full column tile: unmasked inner loop, pointer-bumped B --
        const float* wp = W + (size_t)(hi << 1) * FOUT + ncol;
        #pragma unroll 4
        for (int k0 = 0; k0 < FIN; k0 += 4) {
            // A 16x4 fragment: lane holds X[m][k0+2*hi], X[m][k0+2*hi+1]
            v2f a = *reinterpret_cast<const v2f*>(xrow + k0);
            // B 4x16 fragment: lane holds W[k0+2*hi][ncol], W[k0+2*hi+1][ncol]
            v2f b;
            b.x = wp[0];
            b.y = wp[FOUT];
            wp += 4 * FOUT;
            acc = __builtin_amdgcn_wmma_f32_16x16x4_f32(false, a, false, b,
                                                        (short)0, acc, false, false);
        }
    } else {
        // -------- partial tail tile (layer 1, cols 32..39): masked loads ---
        const int ncolc  = (ncol < FOUT) ? ncol : (FOUT - 1);
        const float bmask = (ncol < FOUT) ? 1.0f : 0.0f;
        const float* wp = W + (size_t)(hi << 1) * FOUT + ncolc;
        #pragma unroll 4
        for (int k0 = 0; k0 < FIN; k0 += 4) {
            v2f a = *reinterpret_cast<const v2f*>(xrow + k0);
            v2f b;
            b.x = wp[0]    * bmask;
            b.y = wp[FOUT] * bmask;
            wp += 4 * FOUT;
            acc = __builtin_amdgcn_wmma_f32_16x16x4_f32(false, a, false, b,
                                                        (short)0, acc, false, false);
        }
    }

    if (ncol < FOUT) {
        const int mo = rowBase + (hi << 3);     // lanes 16-31 hold M = r+8
        #pragma unroll
        for (int r = 0; r < 8; ++r)
            O[(size_t)(mo + r) * FOUT + ncol] = acc[r];
    }
}

// -------------------------------------------------- edge scatter: agg[row] += src[col]
template<int F>
__global__ void scatter_add(const float* __restrict__ src,
                            const int* __restrict__ rows, const int* __restrict__ cols,
                            float* __restrict__ agg, int E)
{
    constexpr int F4 = F >> 2;
    long long t = (long long)blockIdx.x * blockDim.x + threadIdx.x;
    if (t >= (long long)E * F4) return;
    int e = (int)(t / F4);                      // constant divisor -> mul/shift
    int c = ((int)(t % F4)) << 2;
    int r  = rows[e];
    int cl = cols[e];
    const float4 v = *reinterpret_cast<const float4*>(src + (size_t)cl * F + c);
    float* dst = agg + (size_t)r * F + c;
    __hip_atomic_fetch_add(dst + 0, v.x, __ATOMIC_RELAXED, __HIP_MEMORY_SCOPE_AGENT);
    __hip_atomic_fetch_add(dst + 1, v.y, __ATOMIC_RELAXED, __HIP_MEMORY_SCOPE_AGENT);
    __hip_atomic_fetch_add(dst + 2, v.z, __ATOMIC_RELAXED, __HIP_MEMORY_SCOPE_AGENT);
    __hip_atomic_fetch_add(dst + 3, v.w, __ATOMIC_RELAXED, __HIP_MEMORY_SCOPE_AGENT);
}

// mode 0: M = relu(d_inv*(M+agg))           (out_low)
// mode 1: M = relu(M - d_inv*(M+agg))       (out_high)
template<int F>
__global__ void lowhigh_kernel(float* __restrict__ M, const float* __restrict__ agg,
                               const float* __restrict__ dinv,
                               long long total, int mode)
{
    long long idx = (long long)blockIdx.x * blockDim.x + threadIdx.x;
    if (idx >= total) return;
    int i = (int)(idx / F);                     // constant divisor -> mul/shift
    float mv  = M[idx];
    float low = dinv[i] * (mv + agg[idx]);
    float v   = (mode == 0) ? low : (mv - low);
    M[idx] = fmaxf(v, 0.0f);
}

// ---------------------------------------- per-node attention mix (one wave/node)
template<int F>
__global__ void combine_kernel(const float* __restrict__ outL, const float* __restrict__ outH,
                               const float* __restrict__ hm,
                               const float* __restrict__ al, const float* __restrict__ ah,
                               const float* __restrict__ am, const float* __restrict__ att,
                               float* __restrict__ out, int n)
{
    int wave = threadIdx.x >> 5;
    int lane = threadIdx.x & 31;
    int node = blockIdx.x * (blockDim.x >> 5) + wave;
    if (node >= n) return;                       // whole-wave uniform

    const float* pl = outL + (size_t)node * F;
    const float* ph = outH + (size_t)node * F;
    const float* pm = hm   + (size_t)node * F;

    float sl = 0.f, sh = 0.f, sm = 0.f;
    #pragma unroll
    for (int f0 = 0; f0 < F; f0 += 32) {
        int f = f0 + lane;
        if (f < F) {
            sl += pl[f] * al[f];
            sh += ph[f] * ah[f];
            sm += fmaxf(pm[f], 0.f) * am[f];
        }
    }
    #pragma unroll
    for (int o = 16; o >= 1; o >>= 1) {          // wave32 butterfly reduce
        sl += __shfl_xor(sl, o, 32);
        sh += __shfl_xor(sh, o, 32);
        sm += __shfl_xor(sm, o, 32);
    }

    float g0 = 1.f / (1.f + expf(-sl));
    float g1 = 1.f / (1.f + expf(-sh));
    float g2 = 1.f / (1.f + expf(-sm));
    const float T = 1.0f / 3.0f;                 // softmax((sig@att)/3)
    float t0 = (g0 * att[0] + g1 * att[3] + g2 * att[6]) * T;
    float t1 = (g0 * att[1] + g1 * att[4] + g2 * att[7]) * T;
    float t2 = (g0 * att[2] + g1 * att[5] + g2 * att[8]) * T;
    float mx = fmaxf(t0, fmaxf(t1, t2));
    float e0 = expf(t0 - mx), e1 = expf(t1 - mx), e2 = expf(t2 - mx);
    float inv = 3.0f / (e0 + e1 + e2);           // fold final *3.0 in here
    float a0 = e0 * inv, a1 = e1 * inv, a2 = e2 * inv;

    float* po = out + (size_t)node * F;
    #pragma unroll
    for (int f0 = 0; f0 < F; f0 += 32) {
        int f = f0 + lane;
        if (f < F)
            po[f] = a0 * pl[f] + a1 * ph[f] + a2 * fmaxf(pm[f], 0.f);
    }
}

// --------------------------------------------------------------- launcher
static inline unsigned cdiv(long long a, long long b) { return (unsigned)((a + b - 1) / b); }

extern "C" void kernel_launch(void* const* d_in, const int* in_sizes, int n_in,
                              void* d_out, int out_size, void* d_ws, size_t ws_size,
                              hipStream_t stream)
{
    const float* x    = (const float*)d_in[0];
    const int*   ei   = (const int*)d_in[1];          // jax default x64-off -> int32
    const float* Wl0  = (const float*)d_in[2];
    const float* Wh0  = (const float*)d_in[3];
    const float* Wm0  = (const float*)d_in[4];
    const float* al0  = (const float*)d_in[5];
    const float* ah0  = (const float*)d_in[6];
    const float* am0  = (const float*)d_in[7];
    const float* att0 = (const float*)d_in[8];
    const float* Wl1  = (const float*)d_in[9];
    const float* Wh1  = (const float*)d_in[10];
    const float* Wm1  = (const float*)d_in[11];
    const float* al1  = (const float*)d_in[12];
    const float* ah1  = (const float*)d_in[13];
    const float* am1  = (const float*)d_in[14];
    const float* att1 = (const float*)d_in[15];
    float* out = (float*)d_out;

    const int N = N_NODES, E = N_EDGES;
    const int* rows = ei;
    const int* cols = ei + E;

    // workspace layout (floats): dinv | B0 | B1 | B2 | AGG | H   (~128 MB)
    float* ws   = (float*)d_ws;
    float* dinv = ws;
    float* B0   = dinv + N;
    float* B1   = B0 + (size_t)N * HID_F;
    float* B2   = B1 + (size_t)N * HID_F;
    float* AGG  = B2 + (size_t)N * HID_F;
    float* H    = AGG + (size_t)N * HID_F;

    // degree -> d_inv
    zero_f32<<<cdiv(N, 256), 256, 0, stream>>>(dinv, N);
    degree_kernel<<<cdiv(E, 256), 256, 0, stream>>>(rows, dinv, E);
    dinv_kernel<<<cdiv(N, 256), 256, 0, stream>>>(dinv, N);

    // ===================== layer 0 (256 -> 128) =====================
    gemm3_wmma<IN_F, HID_F><<<dim3(N / 16, 3), 256, 0, stream>>>(x, Wl0, Wh0, Wm0,
                                                                 B0, B1, B2);
    const long long tot0 = (long long)N * HID_F;
    const long long st0  = (long long)E * (HID_F / 4);
    zero_f32<<<cdiv(tot0, 256), 256, 0, stream>>>(AGG, tot0);
    scatter_add<HID_F><<<cdiv(st0, 256), 256, 0, stream>>>(B0, rows, cols, AGG, E);
    lowhigh_kernel<HID_F><<<cdiv(tot0, 256), 256, 0, stream>>>(B0, AGG, dinv, tot0, 0);
    zero_f32<<<cdiv(tot0, 256), 256, 0, stream>>>(AGG, tot0);
    scatter_add<HID_F><<<cdiv(st0, 256), 256, 0, stream>>>(B1, rows, cols, AGG, E);
    lowhigh_kernel<HID_F><<<cdiv(tot0, 256), 256, 0, stream>>>(B1, AGG, dinv, tot0, 1);
    combine_kernel<HID_F><<<cdiv(N, 8), 256, 0, stream>>>(B0, B1, B2, al0, ah0, am0, att0,
                                                          H, N);

    // ===================== layer 1 (128 -> 40) ======================
    gemm3_wmma<HID_F, OUT_F><<<dim3(N / 16, 3), 96, 0, stream>>>(H, Wl1, Wh1, Wm1,
                                                                 B0, B1, B2);
    const long long tot1 = (long long)N * OUT_F;
    const long long st1  = (long long)E * (OUT_F / 4);
    zero_f32<<<cdiv(tot1, 256), 256, 0, stream>>>(AGG, tot1);
    scatter_add<OUT_F><<<cdiv(st1, 256), 256, 0, stream>>>(B0, rows, cols, AGG, E);
    lowhigh_kernel<OUT_F><<<cdiv(tot1, 256), 256, 0, stream>>>(B0, AGG, dinv, tot1, 0);
    zero_f32<<<cdiv(tot1, 256), 256, 0, stream>>>(AGG, tot1);
    scatter_add<OUT_F><<<cdiv(st1, 256), 256, 0, stream>>>(B1, rows, cols, AGG, E);
    lowhigh_kernel<OUT_F><<<cdiv(tot1, 256), 256, 0, stream>>>(B1, AGG, dinv, tot1, 1);
    combine_kernel<OUT_F><<<cdiv(N, 8), 256, 0, stream>>>(B0, B1, B2, al1, ah1, am1, att1,
                                                          out, N);
}